// GATblock_58282706206740
// MI455X (gfx1250) — compile-verified
//
#include <hip/hip_runtime.h>

typedef __attribute__((ext_vector_type(2))) float v2f;
typedef __attribute__((ext_vector_type(8))) float v8f;

#define GAT_N      10000
#define NEG_SLOPE  0.2f
#define GAT_EPS    1e-16f
#define ORD_NEG_INF 0x007FFFFFu   // f2ord(-inf)

// ---- order-preserving float <-> uint (for atomic segment-max) ----
__device__ __forceinline__ unsigned f2ord(float f) {
  unsigned u = __float_as_uint(f);
  return (u & 0x80000000u) ? ~u : (u | 0x80000000u);
}
__device__ __forceinline__ float ord2f(unsigned o) {
  unsigned u = (o & 0x80000000u) ? (o ^ 0x80000000u) : ~o;
  return __uint_as_float(u);
}

__global__ void fill_u32(unsigned* __restrict__ p, unsigned v, int n) {
  int i = blockIdx.x * blockDim.x + threadIdx.x;
  if (i < n) p[i] = v;
}

// C[M,N] = A[M,K] * B[K,N]; row-major; M%16==0, N%16==0, K%4==0.
// One wave per 16x16 output tile, V_WMMA_F32_16X16X4_F32 over K.
__global__ void gemm_f32_wmma(const float* __restrict__ A, const float* __restrict__ B,
                              float* __restrict__ C, int M, int N, int K) {
  const int lane  = threadIdx.x & 31;
  const int wave  = threadIdx.x >> 5;
  const int nt_n  = N >> 4;
  const int tile  = blockIdx.x * (blockDim.x >> 5) + wave;
  if (tile >= (M >> 4) * nt_n) return;          // wave-uniform: EXEC stays all-ones
  const int mt    = tile / nt_n;
  const int nt    = tile - mt * nt_n;
  const int l15   = lane & 15;
  const int khalf = (lane >> 4) << 1;           // lanes 0-15 -> K 0,1 ; lanes 16-31 -> K 2,3
  const float* ap = A + (size_t)(mt * 16 + l15) * K + khalf;
  const float* bp = B + (size_t)khalf * N + nt * 16 + l15;
  v8f acc = {};
  for (int k = 0; k < K; k += 4) {
    v2f a, b;
    a.x = ap[0]; a.y = ap[1];                   // contiguous pair -> b64 load
    b.x = bp[0]; b.y = bp[N];
    ap += 4;
    bp += (size_t)4 * N;
    acc = __builtin_amdgcn_wmma_f32_16x16x4_f32(false, a, false, b, (short)0, acc,
                                                false, false);
  }
  // C/D layout: lanes 0-15: M = mt*16 + v ; lanes 16-31: M = mt*16 + 8 + v ; N = l15
  float* cp = C + (size_t)(mt * 16 + ((lane >> 4) << 3)) * N + nt * 16 + l15;
#pragma unroll
  for (int v = 0; v < 8; ++v) cp[(size_t)v * N] = acc[v];
}

// a_src[n,h] = dot(h[n, h*64 : h*64+64], att_src[h]); same for a_dst.
__global__ void att_dots(const float* __restrict__ h, const float* __restrict__ att_s,
                         const float* __restrict__ att_d, int N, int H,
                         float* __restrict__ out_s, float* __restrict__ out_d) {
  int i = blockIdx.x * blockDim.x + threadIdx.x;
  if (i >= N * H) return;
  int n = i / H, hh = i - n * H;
  const float* hp = h + (size_t)n * H * 64 + hh * 64;
  const float* sp = att_s + hh * 64;
  const float* dp = att_d + hh * 64;
  float s1 = 0.f, s2 = 0.f;
#pragma unroll 8
  for (int c = 0; c < 64; ++c) {
    float v = hp[c];
    s1 += v * sp[c];
    s2 += v * dp[c];
  }
  out_s[i] = s1;
  out_d[i] = s2;
}

// thread per (edge, head): score = leaky_relu(a_src[s]+a_dst[d]); segment max over dst.
__global__ void edge_max(const int* __restrict__ src, const int* __restrict__ dst,
                         int E, int N, int H,
                         const float* __restrict__ as_, const float* __restrict__ ad_,
                         unsigned* __restrict__ m) {
  int i = blockIdx.x * blockDim.x + threadIdx.x;
  int ET = E + N;
  if (i >= ET * H) return;
  int e = i / H, h = i - e * H;
  int s = (e < E) ? src[e] : (e - E);           // implicit self-loops
  int d = (e < E) ? dst[e] : (e - E);
  float sc = as_[s * H + h] + ad_[d * H + h];
  sc = (sc > 0.f) ? sc : NEG_SLOPE * sc;
  atomicMax(&m[d * H + h], f2ord(sc));
}

// thread per (edge, head): ex = exp(score - m[dst]); segment sum; stash ex per edge.
__global__ void edge_expsum(const int* __restrict__ src, const int* __restrict__ dst,
                            int E, int N, int H,
                            const float* __restrict__ as_, const float* __restrict__ ad_,
                            const unsigned* __restrict__ m,
                            float* __restrict__ exbuf, float* __restrict__ ssum) {
  int i = blockIdx.x * blockDim.x + threadIdx.x;
  int ET = E + N;
  if (i >= ET * H) return;
  int e = i / H, h = i - e * H;
  int s = (e < E) ? src[e] : (e - E);
  int d = (e < E) ? dst[e] : (e - E);
  float sc = as_[s * H + h] + ad_[d * H + h];
  sc = (sc > 0.f) ? sc : NEG_SLOPE * sc;
  float ex = __expf(sc - ord2f(m[d * H + h]));
  exbuf[i] = ex;
  atomicAdd(&ssum[d * H + h], ex);
}

// wave per edge: agg[dst, :] += alpha_h * h[src, :]  (C = H*64 channels, coalesced)
__global__ void edge_aggregate(const int* __restrict__ src, const int* __restrict__ dst,
                               int E, int N, int H, int C,
                               const float* __restrict__ h,
                               const float* __restrict__ exbuf,
                               const float* __restrict__ ssum,
                               float* __restrict__ agg) {
  int lane = threadIdx.x & 31;
  int wave = threadIdx.x >> 5;
  int e = blockIdx.x * (blockDim.x >> 5) + wave;
  int ET = E + N;
  if (e >= ET) return;
  int s = (e < E) ? src[e] : (e - E);
  int d = (e < E) ? dst[e] : (e - E);
  const float* hp = h + (size_t)s * C;
  float* op = agg + (size_t)d * C;
  for (int c = lane; c < C; c += 32) {
    int hh = c >> 6;
    float alpha = exbuf[e * H + hh] / (ssum[d * H + hh] + GAT_EPS);
    atomicAdd(&op[c], alpha * hp[c]);
  }
}

__global__ void bias_relu(float* __restrict__ x, const float* __restrict__ b,
                          int total, int C) {
  int i = blockIdx.x * blockDim.x + threadIdx.x;
  if (i >= total) return;
  float v = x[i] + b[i % C];
  x[i] = (v > 0.f) ? v : 0.f;
}

static inline int cdiv(int a, int b) { return (a + b - 1) / b; }

extern "C" void kernel_launch(void* const* d_in, const int* in_sizes, int n_in,
                              void* d_out, int out_size, void* d_ws, size_t ws_size,
                              hipStream_t stream) {
  (void)n_in; (void)out_size; (void)ws_size;
  const float* x     = (const float*)d_in[0];
  const int*   ei    = (const int*)d_in[1];
  const float* W1    = (const float*)d_in[2];
  const float* asc1  = (const float*)d_in[3];
  const float* adc1  = (const float*)d_in[4];
  const float* b1    = (const float*)d_in[5];
  const float* W2    = (const float*)d_in[6];
  const float* asc2  = (const float*)d_in[7];
  const float* adc2  = (const float*)d_in[8];
  const float* b2    = (const float*)d_in[9];

  const int N  = GAT_N;
  const int E  = in_sizes[1] / 2;
  const int ET = E + N;
  const int* srcp = ei;
  const int* dstp = ei + E;

  // ---- workspace carve-up (all fits in L2: ~37 MB) ----
  float*    h1   = (float*)d_ws;                 // N*320
  float*    agg1 = h1   + (size_t)N * 320;       // N*320
  float*    as1  = agg1 + (size_t)N * 320;       // N*5
  float*    ad1  = as1  + (size_t)N * 5;         // N*5
  unsigned* m1   = (unsigned*)(ad1 + (size_t)N * 5); // N*5
  float*    s1   = (float*)(m1 + (size_t)N * 5); // N*5
  float*    ex1  = s1   + (size_t)N * 5;         // ET*5
  float*    h2   = ex1  + (size_t)ET * 5;        // N*64
  float*    as2  = h2   + (size_t)N * 64;        // N
  float*    ad2  = as2  + N;                     // N
  unsigned* m2   = (unsigned*)(ad2 + N);         // N
  float*    s2   = (float*)(m2 + N);             // N
  float*    ex2  = s2   + N;                     // ET
  float*    out  = (float*)d_out;                // N*64

  const int TB = 256;

  // ---- re-init accumulators every call (self-contained launch) ----
  fill_u32<<<cdiv(N * 320, TB), TB, 0, stream>>>((unsigned*)agg1, 0u, N * 320);
  fill_u32<<<cdiv(N * 5,   TB), TB, 0, stream>>>((unsigned*)s1,   0u, N * 5);
  fill_u32<<<cdiv(N * 5,   TB), TB, 0, stream>>>(m1, ORD_NEG_INF, N * 5);
  fill_u32<<<cdiv(N * 64,  TB), TB, 0, stream>>>((unsigned*)out,  0u, N * 64);
  fill_u32<<<cdiv(N,       TB), TB, 0, stream>>>((unsigned*)s2,   0u, N);
  fill_u32<<<cdiv(N,       TB), TB, 0, stream>>>(m2, ORD_NEG_INF, N);

  // ================= Layer 1 (heads=5, C=64, concat) =================
  {
    int tiles = (N / 16) * (320 / 16);           // 625 * 20
    gemm_f32_wmma<<<cdiv(tiles * 32, TB), TB, 0, stream>>>(x, W1, h1, N, 320, 128);
  }
  att_dots<<<cdiv(N * 5, TB), TB, 0, stream>>>(h1, asc1, adc1, N, 5, as1, ad1);
  edge_max<<<cdiv(ET * 5, TB), TB, 0, stream>>>(srcp, dstp, E, N, 5, as1, ad1, m1);
  edge_expsum<<<cdiv(ET * 5, TB), TB, 0, stream>>>(srcp, dstp, E, N, 5, as1, ad1, m1, ex1, s1);
  edge_aggregate<<<cdiv(ET, 8), TB, 0, stream>>>(srcp, dstp, E, N, 5, 320, h1, ex1, s1, agg1);
  bias_relu<<<cdiv(N * 320, TB), TB, 0, stream>>>(agg1, b1, N * 320, 320);

  // ================= Layer 2 (heads=1, C=64, mean==identity) =========
  {
    int tiles = (N / 16) * (64 / 16);            // 625 * 4
    gemm_f32_wmma<<<cdiv(tiles * 32, TB), TB, 0, stream>>>(agg1, W2, h2, N, 64, 320);
  }
  att_dots<<<cdiv(N, TB), TB, 0, stream>>>(h2, asc2, adc2, N, 1, as2, ad2);
  edge_max<<<cdiv(ET, TB), TB, 0, stream>>>(srcp, dstp, E, N, 1, as2, ad2, m2);
  edge_expsum<<<cdiv(ET, TB), TB, 0, stream>>>(srcp, dstp, E, N, 1, as2, ad2, m2, ex2, s2);
  edge_aggregate<<<cdiv(ET, 8), TB, 0, stream>>>(srcp, dstp, E, N, 1, 64, h2, ex2, s2, out);
  bias_relu<<<cdiv(N * 64, TB), TB, 0, stream>>>(out, b2, N * 64, 64);
}